// TripletMarginMiner_79250736546712
// MI455X (gfx1250) — compile-verified
//
#include <hip/hip_runtime.h>
#include <hip/hip_bf16.h>

typedef __attribute__((ext_vector_type(2))) float v2f;
typedef __attribute__((ext_vector_type(4))) float v4f;
typedef __attribute__((ext_vector_type(8))) float v8f;

#define N      512
#define D      256
#define MARGIN 0.2f

// ---------------------------------------------------------------------------
// Kernel 1: squared L2 norm of each embedding row.  512 threads total.
// ---------------------------------------------------------------------------
__global__ void sq_kernel(const float* __restrict__ emb, float* __restrict__ sq) {
    int r = blockIdx.x * blockDim.x + threadIdx.x;   // 0..511
    const v4f* row = (const v4f*)(emb + (size_t)r * D);
    float s = 0.0f;
#pragma unroll 8
    for (int i = 0; i < D / 4; ++i) {
        v4f v = row[i];
        s += v.x * v.x + v.y * v.y + v.z * v.z + v.w * v.w;
    }
    sq[r] = s;
}

// ---------------------------------------------------------------------------
// Kernel 2: Gram matrix via V_WMMA_F32_16X16X4_F32, fused into the clamped
// squared-distance matrix  mat[m][n] = max(0, sq[m] + sq[n] - 2*<e_m,e_n>).
// One wave32 per 16x16 output tile; K=256 in steps of 4.
//
// A (16x4 f32) layout per ISA: lanes 0-15 hold M=lane, K={0,1} in VGPR0/1;
// lanes 16-31 hold M=lane-16, K={2,3}.  B = A^T of the same matrix, so the
// per-lane load pattern is identical with N in place of M.
// ---------------------------------------------------------------------------
__global__ void gram_kernel(const float* __restrict__ emb,
                            const float* __restrict__ sq,
                            float* __restrict__ mat) {
    const int lane = threadIdx.x;        // wave32
    const int half = lane >> 4;          // 0: K pair {0,1}, 1: K pair {2,3}
    const int idx  = lane & 15;

    const int mrow = blockIdx.y * 16 + idx;   // A-matrix row for this lane
    const int nrow = blockIdx.x * 16 + idx;   // B-matrix col (= emb row)

    const v2f* Arow = (const v2f*)(emb + (size_t)mrow * D);
    const v2f* Brow = (const v2f*)(emb + (size_t)nrow * D);

    v8f acc = {};
#pragma unroll 4
    for (int i = 0; i < D / 4; ++i) {             // 64 WMMAs, K step 4
        v2f a = Arow[2 * i + half];
        v2f b = Brow[2 * i + half];
        // 8 args: (neg_a, A, neg_b, B, c_mod, C, reuse_a, reuse_b)
        acc = __builtin_amdgcn_wmma_f32_16x16x4_f32(
            false, a, false, b, (short)0, acc, false, false);
    }

    // C/D layout: VGPR v -> row M = v + 8*half, col N = lane&15.
    const int n  = blockIdx.x * 16 + idx;
    const float sn = sq[n];
#pragma unroll
    for (int v = 0; v < 8; ++v) {
        int m = blockIdx.y * 16 + v + 8 * half;
        float d = fmaxf(sq[m] + sn - 2.0f * acc[v], 0.0f);
        mat[(size_t)m * N + n] = d;
    }
}

// ---------------------------------------------------------------------------
// Kernel 3: mine the [a,p,n] mask.  One block per anchor a (512 blocks x 256
// threads).  Store-bandwidth bound: 512^3 floats = 536MB, streamed with
// nontemporal 128-bit stores (output >> L2, NT avoids thrashing).
//
// nm[n] = mat[a,n] if label[n] != label[a] else +INF, so the per-element test
// collapses to  nm[n] <= mat[a,p] + margin  (gated by the (a,p) validity).
// Each thread's float4 of nm is p-invariant -> held in registers.
// ---------------------------------------------------------------------------
__global__ void mask_kernel(const float* __restrict__ mat,
                            const int* __restrict__ labels,
                            float* __restrict__ out) {
    __shared__ float s_mat[N];
    __shared__ float s_nm[N];
    __shared__ int   s_lab[N];

    const int tid = threadIdx.x;          // 0..255
    const int a   = blockIdx.x;           // 0..511

    for (int i = tid; i < N; i += 256) {
        s_mat[i] = mat[(size_t)a * N + i];
        s_lab[i] = labels[i];
    }
    __syncthreads();

    const int la = s_lab[a];
    for (int i = tid; i < N; i += 256)
        s_nm[i] = (s_lab[i] != la) ? s_mat[i] : __builtin_inff();
    __syncthreads();

    const int n4   = tid & 127;           // which float4 of the n-row
    const int psub = tid >> 7;            // 0 or 1: which p of the pair
    const v4f nm   = ((const v4f*)s_nm)[n4];   // p-invariant, in registers

    float* base = out + (((size_t)a) << 18);   // a * 512 * 512

    for (int pb = 0; pb < N; pb += 2) {
        const int  p    = pb + psub;
        const bool vap  = (s_lab[p] == la) & (p != a);
        const float thr = s_mat[p] + MARGIN;

        v4f r;
        r.x = (vap && nm.x <= thr) ? 1.0f : 0.0f;
        r.y = (vap && nm.y <= thr) ? 1.0f : 0.0f;
        r.z = (vap && nm.z <= thr) ? 1.0f : 0.0f;
        r.w = (vap && nm.w <= thr) ? 1.0f : 0.0f;

        v4f* dst = (v4f*)(base + ((size_t)p << 9) + (n4 << 2));
        __builtin_nontemporal_store(r, dst);
    }
}

// ---------------------------------------------------------------------------
extern "C" void kernel_launch(void* const* d_in, const int* in_sizes, int n_in,
                              void* d_out, int out_size, void* d_ws, size_t ws_size,
                              hipStream_t stream) {
    const float* emb    = (const float*)d_in[0];   // [512, 256] f32
    const int*   labels = (const int*)d_in[1];     // [512] int
    float*       out    = (float*)d_out;           // [512,512,512] mask as f32

    float* sq  = (float*)d_ws;                     // 512 floats
    float* mat = sq + N;                           // 512*512 floats (1 MB)

    sq_kernel  <<<N / 256, 256, 0, stream>>>(emb, sq);
    gram_kernel<<<dim3(N / 16, N / 16), 32, 0, stream>>>(emb, sq, mat);
    mask_kernel<<<N, 256, 0, stream>>>(mat, labels, out);
}